// DispatchBiasCombine_62878321214337
// MI455X (gfx1250) — compile-verified
//
#include <hip/hip_runtime.h>
#include <hip/hip_bf16.h>

typedef float v2f __attribute__((ext_vector_type(2)));
typedef float v4f __attribute__((ext_vector_type(4)));
typedef float v8f __attribute__((ext_vector_type(8)));

#define NEXP 64
#define KTOP 8

// ---------------------------------------------------------------------------
// Kernel 1: logits[T,64] = flat[T,H] @ W[64,H]^T + rbias, via
// V_WMMA_F32_16X16X4_F32. One wave handles 16 tokens x all 64 experts.
// A lane layout (16x4 f32): lane l<16 holds M=l,K={k,k+1}; lane 16+l holds
// M=l,K={k+2,k+3}  -> contiguous float2 along H. B (4x16 = W^T tile) is the
// same pattern on W rows. C layout: VGPR g = row 8*half+g, col = lane&15.
//
// Explicit double buffering: two disjoint register sets (no inter-stage
// copies). While stage S's 4 WMMAs execute, stage S^1's 5 loads are in
// flight, so the compiler only needs s_wait_loadcnt <= 5 instead of 0.
// ---------------------------------------------------------------------------
__global__ __launch_bounds__(128) void router_gemm_wmma(
    const float* __restrict__ flat,   // [T,H]  (streamed, non-temporal)
    const float* __restrict__ W,      // [E,H]  (hot, keep cached)
    const float* __restrict__ rbias,  // [E]
    float* __restrict__ logits,       // [T,E]
    int H)
{
  const int lane = threadIdx.x & 31;
  const int wave = threadIdx.x >> 5;
  const int l    = lane & 15;
  const int half = lane >> 4;
  const int tileM = (blockIdx.x * 4 + wave) * 16;

  const float* aRow  = flat + (size_t)(tileM + l) * H + 2 * half;
  const float* bRow0 = W + (size_t)(l +  0) * H + 2 * half;
  const float* bRow1 = W + (size_t)(l + 16) * H + 2 * half;
  const float* bRow2 = W + (size_t)(l + 32) * H + 2 * half;
  const float* bRow3 = W + (size_t)(l + 48) * H + 2 * half;

  v8f acc0 = {}, acc1 = {}, acc2 = {}, acc3 = {};

  // Stage buffers: two fully independent register sets.
  v2f a0, p00, p01, p02, p03;   // buffer 0
  v2f a1, p10, p11, p12, p13;   // buffer 1

  // Preload k=0 into buf0, k=4 into buf1.
  a0  = __builtin_nontemporal_load((const v2f*)(aRow));
  p00 = *(const v2f*)(bRow0);
  p01 = *(const v2f*)(bRow1);
  p02 = *(const v2f*)(bRow2);
  p03 = *(const v2f*)(bRow3);

  a1  = __builtin_nontemporal_load((const v2f*)(aRow + 4));
  p10 = *(const v2f*)(bRow0 + 4);
  p11 = *(const v2f*)(bRow1 + 4);
  p12 = *(const v2f*)(bRow2 + 4);
  p13 = *(const v2f*)(bRow3 + 4);

  for (int k = 8; k <= H - 8; k += 8) {
    // Consume buf0 (waits only for buf0's loads; buf1's stay in flight),
    // then immediately refill buf0 for step k.
    acc0 = __builtin_amdgcn_wmma_f32_16x16x4_f32(false, a0, false, p00, (short)0, acc0, false, false);
    acc1 = __builtin_amdgcn_wmma_f32_16x16x4_f32(false, a0, false, p01, (short)0, acc1, false, false);
    acc2 = __builtin_amdgcn_wmma_f32_16x16x4_f32(false, a0, false, p02, (short)0, acc2, false, false);
    acc3 = __builtin_amdgcn_wmma_f32_16x16x4_f32(false, a0, false, p03, (short)0, acc3, false, false);
    a0  = __builtin_nontemporal_load((const v2f*)(aRow + k));
    p00 = *(const v2f*)(bRow0 + k);
    p01 = *(const v2f*)(bRow1 + k);
    p02 = *(const v2f*)(bRow2 + k);
    p03 = *(const v2f*)(bRow3 + k);

    // Consume buf1, refill buf1 for step k+4.
    acc0 = __builtin_amdgcn_wmma_f32_16x16x4_f32(false, a1, false, p10, (short)0, acc0, false, false);
    acc1 = __builtin_amdgcn_wmma_f32_16x16x4_f32(false, a1, false, p11, (short)0, acc1, false, false);
    acc2 = __builtin_amdgcn_wmma_f32_16x16x4_f32(false, a1, false, p12, (short)0, acc2, false, false);
    acc3 = __builtin_amdgcn_wmma_f32_16x16x4_f32(false, a1, false, p13, (short)0, acc3, false, false);
    a1  = __builtin_nontemporal_load((const v2f*)(aRow + k + 4));
    p10 = *(const v2f*)(bRow0 + k + 4);
    p11 = *(const v2f*)(bRow1 + k + 4);
    p12 = *(const v2f*)(bRow2 + k + 4);
    p13 = *(const v2f*)(bRow3 + k + 4);
  }

  // Epilogue: drain both buffers.
  acc0 = __builtin_amdgcn_wmma_f32_16x16x4_f32(false, a0, false, p00, (short)0, acc0, false, false);
  acc1 = __builtin_amdgcn_wmma_f32_16x16x4_f32(false, a0, false, p01, (short)0, acc1, false, false);
  acc2 = __builtin_amdgcn_wmma_f32_16x16x4_f32(false, a0, false, p02, (short)0, acc2, false, false);
  acc3 = __builtin_amdgcn_wmma_f32_16x16x4_f32(false, a0, false, p03, (short)0, acc3, false, false);
  acc0 = __builtin_amdgcn_wmma_f32_16x16x4_f32(false, a1, false, p10, (short)0, acc0, false, false);
  acc1 = __builtin_amdgcn_wmma_f32_16x16x4_f32(false, a1, false, p11, (short)0, acc1, false, false);
  acc2 = __builtin_amdgcn_wmma_f32_16x16x4_f32(false, a1, false, p12, (short)0, acc2, false, false);
  acc3 = __builtin_amdgcn_wmma_f32_16x16x4_f32(false, a1, false, p13, (short)0, acc3, false, false);

  const float rb0 = rbias[l], rb1 = rbias[l + 16], rb2 = rbias[l + 32], rb3 = rbias[l + 48];
  #pragma unroll
  for (int g = 0; g < 8; ++g) {
    const int row = tileM + 8 * half + g;
    float* o = logits + (size_t)row * NEXP;
    o[l]      = acc0[g] + rb0;
    o[l + 16] = acc1[g] + rb1;
    o[l + 32] = acc2[g] + rb2;
    o[l + 48] = acc3[g] + rb3;
  }
}

// ---------------------------------------------------------------------------
// Kernel 2: per-token softmax over 64 logits + top-8 selection.
// One wave32 per token; each lane owns logits[lane] and logits[lane+32].
// Butterfly shfl_xor reductions; argmax tie-break = lower index (matches
// jax.lax.top_k). Writes topk weights (softmax scores), indices, and w_sum.
// ---------------------------------------------------------------------------
__global__ __launch_bounds__(256) void softmax_top8(
    const float* __restrict__ logits,  // [T,E]
    float* __restrict__ topk_w,        // [T,8]
    int*   __restrict__ topk_idx,      // [T,8]
    float* __restrict__ wsum,          // [T]
    int T)
{
  const int lane = threadIdx.x & 31;
  const int wave = threadIdx.x >> 5;
  const int t = blockIdx.x * 8 + wave;
  if (t >= T) return;

  const float l0 = logits[(size_t)t * NEXP + lane];
  const float l1 = logits[(size_t)t * NEXP + 32 + lane];

  float m = fmaxf(l0, l1);
  #pragma unroll
  for (int off = 16; off >= 1; off >>= 1)
    m = fmaxf(m, __shfl_xor(m, off, 32));

  const float e0 = __expf(l0 - m);
  const float e1 = __expf(l1 - m);
  float s = e0 + e1;
  #pragma unroll
  for (int off = 16; off >= 1; off >>= 1)
    s += __shfl_xor(s, off, 32);

  float s0 = e0 / s;
  float s1 = e1 / s;

  float w[KTOP];
  int   id[KTOP];
  float ws_ = 0.0f;

  #pragma unroll
  for (int k = 0; k < KTOP; ++k) {
    float v; int i;
    if (s0 >= s1) { v = s0; i = lane; } else { v = s1; i = lane + 32; }
    #pragma unroll
    for (int off = 16; off >= 1; off >>= 1) {
      float ov = __shfl_xor(v, off, 32);
      int   oi = __shfl_xor(i, off, 32);
      if (ov > v || (ov == v && oi < i)) { v = ov; i = oi; }
    }
    w[k] = v; id[k] = i; ws_ += v;
    if (i == lane)           s0 = -__builtin_huge_valf();
    else if (i == lane + 32) s1 = -__builtin_huge_valf();
  }

  if (lane == 0) {
    #pragma unroll
    for (int k = 0; k < KTOP; ++k) {
      topk_w[(size_t)t * KTOP + k]   = w[k];
      topk_idx[(size_t)t * KTOP + k] = id[k];
    }
    wsum[t] = ws_;
  }
}

// ---------------------------------------------------------------------------
// Kernel 3: out[t,h] = flat[t,h]*wsum[t] + sum_k w_k * bias[idx_k, h]
// One block per token; float4 streaming; per-token scalars staged in LDS.
// flat loads + out stores are non-temporal (pure stream, 512 MB) so the 64
// expert-bias rows (1 MB, re-read by every block) stay L2-resident.
// ---------------------------------------------------------------------------
__global__ __launch_bounds__(256) void combine_kernel(
    const float* __restrict__ flat,      // [T,H]
    const float* __restrict__ ebias,     // [E,H]
    const float* __restrict__ topk_w,    // [T,8]
    const int*   __restrict__ topk_idx,  // [T,8]
    const float* __restrict__ wsum,      // [T]
    float* __restrict__ out,             // [T,H]
    int H)
{
  __shared__ float sw[KTOP];
  __shared__ int   sid[KTOP];
  __shared__ float sws;

  const int t = blockIdx.x;
  if (threadIdx.x < KTOP) {
    sw[threadIdx.x]  = topk_w[(size_t)t * KTOP + threadIdx.x];
    sid[threadIdx.x] = topk_idx[(size_t)t * KTOP + threadIdx.x];
  }
  if (threadIdx.x == 0) sws = wsum[t];
  __syncthreads();

  const float wsv = sws;
  const float w0 = sw[0], w1 = sw[1], w2 = sw[2], w3 = sw[3];
  const float w4 = sw[4], w5 = sw[5], w6 = sw[6], w7 = sw[7];
  const float* b0 = ebias + (size_t)sid[0] * H;
  const float* b1 = ebias + (size_t)sid[1] * H;
  const float* b2 = ebias + (size_t)sid[2] * H;
  const float* b3 = ebias + (size_t)sid[3] * H;
  const float* b4 = ebias + (size_t)sid[4] * H;
  const float* b5 = ebias + (size_t)sid[5] * H;
  const float* b6 = ebias + (size_t)sid[6] * H;
  const float* b7 = ebias + (size_t)sid[7] * H;

  const float* xr   = flat + (size_t)t * H;
  float*       orow = out + (size_t)t * H;

  for (int h = threadIdx.x * 4; h < H; h += 256 * 4) {
    v4f x = __builtin_nontemporal_load((const v4f*)(xr + h));
    v4f acc = x * wsv;
    acc += w0 * (*(const v4f*)(b0 + h));
    acc += w1 * (*(const v4f*)(b1 + h));
    acc += w2 * (*(const v4f*)(b2 + h));
    acc += w3 * (*(const v4f*)(b3 + h));
    acc += w4 * (*(const v4f*)(b4 + h));
    acc += w5 * (*(const v4f*)(b5 + h));
    acc += w6 * (*(const v4f*)(b6 + h));
    acc += w7 * (*(const v4f*)(b7 + h));
    __builtin_nontemporal_store(acc, (v4f*)(orow + h));
  }
}

// ---------------------------------------------------------------------------
extern "C" void kernel_launch(void* const* d_in, const int* in_sizes, int n_in,
                              void* d_out, int out_size, void* d_ws, size_t ws_size,
                              hipStream_t stream) {
  const float* hidden = (const float*)d_in[0];  // [B,S,H] -> flat [T,H]
  const float* rw     = (const float*)d_in[1];  // [E,H]
  const float* rb     = (const float*)d_in[2];  // [E]
  const float* eb     = (const float*)d_in[3];  // [E,H]
  float* out = (float*)d_out;

  const int E = in_sizes[2];                    // 64
  const int H = in_sizes[1] / E;                // 4096
  const int T = in_sizes[0] / H;                // 16384
  (void)E; (void)n_in; (void)out_size; (void)ws_size;

  // Workspace layout
  float* logits   = (float*)d_ws;               // T*64 floats (4 MB)
  float* topk_w   = logits + (size_t)T * NEXP;  // T*8 floats
  int*   topk_idx = (int*)(topk_w + (size_t)T * KTOP);
  float* wsum     = (float*)(topk_idx + (size_t)T * KTOP);

  // 1) Router GEMM via f32 WMMA: 4 waves/block, 16 tokens/wave -> 64 tok/block
  router_gemm_wmma<<<T / 64, 128, 0, stream>>>(hidden, rw, rb, logits, H);

  // 2) Softmax + top-8: 8 tokens per 256-thread block
  softmax_top8<<<(T + 7) / 8, 256, 0, stream>>>(logits, topk_w, topk_idx, wsum, T);

  // 3) Combine: one block per token
  combine_kernel<<<T, 256, 0, stream>>>(hidden, eb, topk_w, topk_idx, wsum, out, H);
}